// Time2dForward_20641612825143
// MI455X (gfx1250) — compile-verified
//
#include <hip/hip_runtime.h>

typedef __attribute__((ext_vector_type(2))) float v2f;
typedef __attribute__((ext_vector_type(8))) float v8f;

#if defined(__has_builtin)
#if __has_builtin(__builtin_amdgcn_global_load_async_to_lds_b128) && \
    __has_builtin(__builtin_amdgcn_s_wait_asynccnt)
#define WAVE_USE_ASYNC_LDS 1
#endif
#endif

#ifdef WAVE_USE_ASYNC_LDS
// Builtin prototype (from hipcc diagnostic): pointee is a 16-B int vector,
// src in addrspace(1) (global), dst in addrspace(3) (LDS).
typedef __attribute__((__vector_size__(4 * sizeof(int)))) int v4i;
typedef __attribute__((address_space(1))) v4i as1_v4i;
typedef __attribute__((address_space(3))) v4i as3_v4i;
#endif

namespace {
constexpr int NXc = 384;
constexpr int NYc = 384;
constexpr int NTc = 600;
constexpr int NEc = 4;
constexpr int DIMc = NXc + 2 * NEc;   // 392 (padded field dimension)
constexpr float DTc = 0.001f;
constexpr float Hc  = 10.0f;
// LDS tile stride: 44 floats = 176 B per row -> 16-B aligned rows for B128
// async writes; gcd(44,64)=4 with period 16 => a 16-lane column access still
// touches 16 distinct banks (conflict-free).
constexpr int LDSW = 44;
}

// Init: edge-padded v^2*dt^2, zero both wavefield buffers, and build the
// banded-stencil WMMA operand table st[6][2][32] (chunk, vgpr, lane) laid out
// exactly as V_WMMA_F32_16X16X4_F32 expects for both the A-side (x-direction)
// and B-side (y-direction) stencil matrices (identical by band symmetry).
__global__ void wave_init(const float* __restrict__ vel,
                          float* __restrict__ v2,
                          float* __restrict__ b0,
                          float* __restrict__ b1,
                          float* __restrict__ st) {
  int id = blockIdx.x * blockDim.x + threadIdx.x;
  if (id < DIMc * DIMc) {
    int i = id / DIMc, j = id % DIMc;
    int vi = min(max(i - NEc, 0), NXc - 1);   // edge-pad (mode='edge')
    int vj = min(max(j - NEc, 0), NYc - 1);
    float v = vel[vi * NYc + vj];
    v2[id] = v * v * DTc * DTc;
    b0[id] = 0.0f;
    b1[id] = 0.0f;
  }
  if (id < 6 * 2 * 32) {
    int c = id >> 6;           // K-chunk 0..5
    int q = (id >> 5) & 1;     // which of the 2 VGPRs
    int l = id & 31;           // lane
    int kb = (l >> 4) << 1;    // lanes 16..31 hold K+2
    int d = 4 * c + kb + q - (l & 15);   // band offset, valid in [0,8]
    float val = 0.0f;
    if (d >= 0 && d <= 8) {
      int k = d - 4; k = k < 0 ? -k : k;
      val = (k == 0) ? (-205.0f / 72.0f)
          : (k == 1) ? (8.0f / 5.0f)
          : (k == 2) ? (-1.0f / 5.0f)
          : (k == 3) ? (8.0f / 315.0f)
          :            (-1.0f / 560.0f);
    }
    st[id] = val;
  }
}

// One leapfrog step. Block = 128 threads = 4 waves, block tile = 32x32 output,
// each wave computes one 16x16 tile with 12 chained v_wmma_f32_16x16x4_f32:
//   lap_x = S(16x24) x Ux(24x16),  lap_y = Uy(16x24) x S'(24x16)
// u13 holds u1 on entry and is overwritten with u3 (each point only reads u1
// at itself, so in-place ping-pong with 2 buffers is exact).
__global__ __launch_bounds__(128) void wave_step(
    float* __restrict__ u13,
    const float* __restrict__ u2,
    const float* __restrict__ v2,
    const float* __restrict__ st,
    const float* __restrict__ w, int t,
    const int* __restrict__ sxp, const int* __restrict__ syp,
    const int* __restrict__ ryp,
    float* __restrict__ frd) {
  __shared__ float s_u[40 * LDSW];

  const int tid = threadIdx.x;
  const int X0 = blockIdx.x * 32;   // interior-coordinate tile origin
  const int Y0 = blockIdx.y * 32;

  // Stage 40x40 halo tile of u2 (padded coords X0..X0+39, Y0..Y0+39).
#ifdef WAVE_USE_ASYNC_LDS
  // Async global->LDS DMA: 10 x B128 per row, no VGPR round-trip (ASYNCcnt).
  for (int idx = tid; idx < 40 * 10; idx += 128) {
    int r  = idx / 10;
    int cq = (idx - r * 10) * 4;                // 16-B-aligned column chunk
    const float* gp = u2 + (X0 + r) * DIMc + (Y0 + cq);
    __builtin_amdgcn_global_load_async_to_lds_b128(
        (as1_v4i*)gp, (as3_v4i*)&s_u[r * LDSW + cq], 0, 0);
  }
  __builtin_amdgcn_s_wait_asynccnt(0);          // this wave's LDS writes done
#else
  for (int idx = tid; idx < 40 * 40; idx += 128) {
    int r = idx / 40, c = idx - r * 40;
    s_u[r * LDSW + c] = u2[(X0 + r) * DIMc + (Y0 + c)];
  }
#endif
  __syncthreads();

  const int lane = tid & 31;
  const int wv   = tid >> 5;
  const int lx0  = (wv & 1) * 16;    // subtile offset in block tile
  const int ly0  = (wv >> 1) * 16;
  const int n    = lane & 15;        // N (and M for A-operand) index
  const int kb   = (lane >> 4) << 1; // upper half-wave holds K+2

  // Stencil operands (same per-lane values serve A-side x and B-side y).
  v2f stc[6];
#pragma unroll
  for (int c = 0; c < 6; ++c) {
    stc[c].x = st[c * 64 + lane];
    stc[c].y = st[c * 64 + 32 + lane];
  }

  v8f acc = {};
  // X direction: A = stencil chunk, B = Ux chunk (rows X0+lx0 .. +23 in LDS).
#pragma unroll
  for (int c = 0; c < 6; ++c) {
    int r0 = lx0 + 4 * c + kb;
    v2f b;
    b.x = s_u[(r0    ) * LDSW + ly0 + 4 + n];
    b.y = s_u[(r0 + 1) * LDSW + ly0 + 4 + n];
    acc = __builtin_amdgcn_wmma_f32_16x16x4_f32(false, stc[c], false, b,
                                                (short)0, acc, false, false);
  }
  // Y direction: A = Uy chunk, B = stencil chunk.
#pragma unroll
  for (int c = 0; c < 6; ++c) {
    int c0 = ly0 + 4 * c + kb;
    v2f a;
    a.x = s_u[(lx0 + 4 + n) * LDSW + c0    ];
    a.y = s_u[(lx0 + 4 + n) * LDSW + c0 + 1];
    acc = __builtin_amdgcn_wmma_f32_16x16x4_f32(false, a, false, stc[c],
                                                (short)0, acc, false, false);
  }

  // D layout: VGPR v, lanes 0-15 -> M=v, lanes 16-31 -> M=v+8; N = lane&15.
  const float h2inv = 1.0f / (Hc * Hc);
  const int sx = *sxp, sy = *syp, ry = *ryp;
#pragma unroll
  for (int v = 0; v < 8; ++v) {
    int mo = v + ((lane >> 4) << 3);
    int gr = X0 + lx0 + mo;          // interior row 0..383
    int gc = Y0 + ly0 + n;           // interior col 0..383
    int gidx = (gr + NEc) * DIMc + (gc + NEc);
    float u2c = s_u[(lx0 + 4 + mo) * LDSW + ly0 + 4 + n];
    float vv  = v2[gidx];
    float val = 2.0f * u2c - u13[gidx] + vv * (acc[v] * h2inv);
    if (gr == sx && gc == sy) val += w[t] * vv;   // source injection
    u13[gidx] = val;                               // u3 written over u1
    if (gc == ry) frd[t * NXc + gr] = val;         // receiver row
  }
}

extern "C" void kernel_launch(void* const* d_in, const int* in_sizes, int n_in,
                              void* d_out, int out_size, void* d_ws, size_t ws_size,
                              hipStream_t stream) {
  const float* vel = (const float*)d_in[0];
  const float* w   = (const float*)d_in[1];
  const int*   sx  = (const int*)d_in[2];
  const int*   sy  = (const int*)d_in[3];
  const int*   ry  = (const int*)d_in[4];
  float* frd = (float*)d_out;

  float* b0 = (float*)d_ws;             // wavefield ping
  float* b1 = b0 + DIMc * DIMc;         // wavefield pong
  float* v2 = b1 + DIMc * DIMc;         // v^2 * dt^2 (padded)
  float* st = v2 + DIMc * DIMc;         // 384-float WMMA stencil table
  // ws needed: (3*392*392 + 384) * 4B ~= 1.85 MB

  const int tot = DIMc * DIMc;
  wave_init<<<(tot + 255) / 256, 256, 0, stream>>>(vel, v2, b0, b1, st);

  for (int t = 0; t < NTc; ++t) {
    float* uprev = (t & 1) ? b1 : b0;   // u1 in, u3 out
    float* ucur  = (t & 1) ? b0 : b1;   // u2
    wave_step<<<dim3(NXc / 32, NYc / 32), 128, 0, stream>>>(
        uprev, ucur, v2, st, w, t, sx, sy, ry, frd);
  }
}